// DeepDFA_64244120813700
// MI455X (gfx1250) — compile-verified
//
#include <hip/hip_runtime.h>
#include <hip/hip_bf16.h>

// DFA scan for MI455X (gfx1250), wave32.
// - One wave per batch chain; 8 waves (256 thr) per workgroup; 128 workgroups.
// - All 32 transition matrices staged in LDS as f16 (256KB of the 320KB/WGP).
// - Transition step: forced v_fma_mix_f32 (f16 LDS operand selected via OPSEL,
//   f32 accumulate), state broadcast via v_readlane -> SGPR operand.
//   4 independent accumulator chains to break FMA latency serialization.
// - Reward projection (shared 64x4 matrix) as real GEMM with
//   V_WMMA_F32_16X16X4_F32 over 16-timestep chunks.

typedef __attribute__((ext_vector_type(2))) float    v2f;
typedef __attribute__((ext_vector_type(8))) float    v8f;
typedef __attribute__((ext_vector_type(2))) _Float16 h2;

#define NA 32
#define NSTATE 64
#define NO 4
#define NB 1024
#define NT 2048

#define WAVES_PER_WG 8
#define THREADS (WAVES_PER_WG * 32)
#define CHUNK 16

// LDS layout (bytes):
//   [0,      262144) : trans matrices f16  [a][s][c], row = 128B (h2-packed cols)
//   [262144, 266240) : Fpad[64][16] f32 (cols 4..15 zero)
//   [266240, 299008) : per-wave state buffer: 16 steps x 64 f32 = 4KB each
#define SM_F_OFF   (262144 / 4)  // float index
#define SM_NS_OFF  (266240 / 4)  // float index
#define SMEM_BYTES 299008

__device__ __forceinline__ float bcast(float v, int l) {
  return __uint_as_float(__builtin_amdgcn_readlane(__float_as_uint(v), l));
}

// acc += f16(lo 16 of m) * s   (f32 fma; MIX sel 2 = {op_sel_hi=1, op_sel=0})
__device__ __forceinline__ void fma_mix_lo(float& acc, unsigned m, float s) {
  asm("v_fma_mix_f32 %0, %1, %2, %0 op_sel:[0,0,0] op_sel_hi:[1,0,0]"
      : "+v"(acc) : "v"(m), "s"(s));
}
// acc += f16(hi 16 of m) * s   (MIX sel 3 = {op_sel_hi=1, op_sel=1})
__device__ __forceinline__ void fma_mix_hi(float& acc, unsigned m, float s) {
  asm("v_fma_mix_f32 %0, %1, %2, %0 op_sel:[1,0,0] op_sel_hi:[1,0,0]"
      : "+v"(acc) : "v"(m), "s"(s));
}

extern "C" __global__ __launch_bounds__(THREADS)
void dfa_scan_kernel(const int* __restrict__ acts,
                     const float* __restrict__ trans,
                     const float* __restrict__ fin,
                     float* __restrict__ out) {
  extern __shared__ char smem_raw[];
  h2*       sM  = (h2*)smem_raw;                 // packed f16 pairs
  unsigned* sMu = (unsigned*)smem_raw;           // same storage, raw b32 view
  float*    sF  = (float*)smem_raw + SM_F_OFF;   // padded reward matrix
  float*    sNS = (float*)smem_raw + SM_NS_OFF;  // per-wave 16x64 state block

  const int tid  = threadIdx.x;
  const int lane = tid & 31;
  const int wave = tid >> 5;

  // ---- stage: trans f32 -> f16 LDS; fin -> zero-padded 64x16 f32 LDS ----
  {
    const float4* t4 = (const float4*)trans;
    for (int i = tid; i < (NA * NSTATE * NSTATE / 4); i += THREADS) {
      float4 v = t4[i];
      h2 p0; p0.x = (_Float16)v.x; p0.y = (_Float16)v.y;
      h2 p1; p1.x = (_Float16)v.z; p1.y = (_Float16)v.w;
      sM[2 * i]     = p0;
      sM[2 * i + 1] = p1;
    }
    for (int i = tid; i < NSTATE * 16; i += THREADS) {
      int r = i >> 4, n = i & 15;
      sF[i] = (n < NO) ? fin[r * NO + n] : 0.0f;
    }
  }
  __syncthreads();

  const int b = blockIdx.x * WAVES_PER_WG + wave;
  float* sNSw = sNS + wave * (CHUNK * NSTATE);

  // lane l holds state elems s[2l] (cur.x) and s[2l+1] (cur.y)
  v2f cur;
  cur.x = (lane == 0) ? 1.0f : 0.0f;
  cur.y = 0.0f;

  const long rew_base = (long)b * NT * NO;
  const int  nl  = lane & 15;           // col/row-within-half for WMMA tiles
  const int  kk  = (lane >> 4) << 1;    // lanes 16-31 own K=2,3 of each tile

  for (int t0 = 0; t0 < NT; t0 += CHUNK) {
    // lanes 0..15 preload the 16 actions of this chunk
    int a_l = acts[(long)b * NT + t0 + nl];

    for (int j = 0; j < CHUNK; ++j) {
      int a     = __builtin_amdgcn_readlane(a_l, j);   // uniform action id
      int aBase = a * (NSTATE * NSTATE / 2);           // b32 index of matrix a

      // 4 independent accumulation chains (even rows / odd rows, col x / col y)
      float ax0 = 0.0f, ax1 = 0.0f, ay0 = 0.0f, ay1 = 0.0f;
      #pragma unroll
      for (int p = 0; p < 32; ++p) {
        float s0 = bcast(cur.x, p);                    // s[2p]
        float s1 = bcast(cur.y, p);                    // s[2p+1]
        unsigned m0 = sMu[aBase + (2 * p) * 32 + lane];     // M[2p  ][2l..2l+1]
        unsigned m1 = sMu[aBase + (2 * p + 1) * 32 + lane]; // M[2p+1][2l..2l+1]
        fma_mix_lo(ax0, m0, s0);   // col 2l   += M[2p  ][2l]   * s[2p]
        fma_mix_hi(ay0, m0, s0);   // col 2l+1 += M[2p  ][2l+1] * s[2p]
        fma_mix_lo(ax1, m1, s1);   // col 2l   += M[2p+1][2l]   * s[2p+1]
        fma_mix_hi(ay1, m1, s1);   // col 2l+1 += M[2p+1][2l+1] * s[2p+1]
      }
      v2f acc; acc.x = ax0 + ax1; acc.y = ay0 + ay1;

      // buffer new state row j for the reward WMMA; advance the chain
      *(v2f*)(sNSw + j * NSTATE + 2 * lane) = acc;
      cur = acc;
    }

    // ---- rewards for 16 steps: (16x64 states) @ (64x16 padded F) ----
    // V_WMMA_F32_16X16X4_F32 layouts (32-bit):
    //   A 16x4: lanes 0-15 -> {K0,K1}, lanes 16-31 -> {K2,K3}, M = lane%16
    //   B 4x16: lanes 0-15 -> rows K0,K1, lanes 16-31 -> rows K2,K3, N = lane%16
    //   C/D 16x16: VGPR r -> rows M=r (lanes 0-15) and M=r+8 (lanes 16-31)
    v8f c = {};
    #pragma unroll
    for (int k = 0; k < 16; ++k) {
      v2f aV = *(const v2f*)(sNSw + nl * NSTATE + 4 * k + kk);
      v2f bV;
      bV.x = sF[(4 * k + kk + 0) * 16 + nl];
      bV.y = sF[(4 * k + kk + 1) * 16 + nl];
      c = __builtin_amdgcn_wmma_f32_16x16x4_f32(false, aV, false, bV,
                                                (short)0, c, false, false);
    }
    if (nl < NO) {                       // only output cols 0..3 are real
      const int tHi = (lane >> 4) << 3;  // +8 rows for upper half lanes
      #pragma unroll
      for (int r = 0; r < 8; ++r) {
        int t = t0 + r + tHi;
        out[rew_base + (long)t * NO + nl] = c[r];
      }
    }
  }

  // final state s_fin[b][2l], s_fin[b][2l+1]
  *(v2f*)(out + (long)NB * NT * NO + (long)b * NSTATE + 2 * lane) = cur;
}

extern "C" void kernel_launch(void* const* d_in, const int* in_sizes, int n_in,
                              void* d_out, int out_size, void* d_ws, size_t ws_size,
                              hipStream_t stream) {
  (void)in_sizes; (void)n_in; (void)d_ws; (void)ws_size; (void)out_size;
  const int*   acts  = (const int*)d_in[0];
  const float* trans = (const float*)d_in[1];
  const float* fin   = (const float*)d_in[2];
  float*       out   = (float*)d_out;

  // CDNA5 allows up to 320KB LDS per workgroup; opt in for 292KB.
  hipFuncSetAttribute((const void*)dfa_scan_kernel,
                      hipFuncAttributeMaxDynamicSharedMemorySize, SMEM_BYTES);

  dim3 grid(NB / WAVES_PER_WG);   // 128 workgroups ~ one per WGP
  dim3 block(THREADS);            // 8 wave32 per workgroup
  dfa_scan_kernel<<<grid, block, SMEM_BYTES, stream>>>(acts, trans, fin, out);
}